// ClassificationHead_67860483277768
// MI455X (gfx1250) — compile-verified
//
#include <hip/hip_runtime.h>
#include <stdint.h>

typedef float v2f __attribute__((ext_vector_type(2)));
typedef float f4  __attribute__((ext_vector_type(4)));
typedef float v8f __attribute__((ext_vector_type(8)));

#define T_    32
#define NWAY  5
#define NSHOT 5
#define NQ    75
#define D_    1024
#define LAMR  100000.0f

#define RCP  1028   // LDS row pitch for Rc / P: 16B aligned, bank stride 4
#define QP   68     // LDS row pitch for staged query tile: 16B aligned, bank stride 4
#define KC   64     // K-chunk per staged tile
#define MT   80     // 75 queries padded to 5 x 16 WMMA M-tiles

__global__ __launch_bounds__(256)
void exemplar_head_kernel(const float* __restrict__ query,
                          const float* __restrict__ support,
                          const float* __restrict__ noise,
                          float* __restrict__ out)
{
    __shared__ __align__(16) float mean_s[D_];
    __shared__ __align__(16) float Rc_s[NSHOT * RCP];
    __shared__ __align__(16) float P_s[NSHOT * RCP];
    __shared__ __align__(16) float Q_s[2][MT * QP];   // double-buffered raw query tiles
    __shared__ float G_s[15];
    __shared__ float M_s[NSHOT][NSHOT];
    __shared__ float pnorm_s[NSHOT];
    __shared__ float mdot_s[NSHOT];
    __shared__ float qnorm_s[MT];

    const int tid  = threadIdx.x;
    const int wave = tid >> 5;
    const int lane = tid & 31;
    const int t = blockIdx.x / NWAY;
    const int w = blockIdx.x % NWAY;

    const float* qbase = query + (size_t)t * NQ * D_;

    // Async tile stager: raw query chunk (80 rows x 64 cols) -> LDS, b128 per lane.
    auto issue_chunk = [&](int buf, int kc) {
        #pragma unroll
        for (int j = 0; j < 5; ++j) {
            int i = tid + j * 256;          // 1280 float4s per tile
            int r = i >> 4;                 // tile row (query)
            int c = (i & 15) * 4;           // column within chunk
            int rr = (r < NQ) ? r : (NQ - 1);   // clamp pad rows (never stored)
            const float* src = qbase + (size_t)rr * D_ + kc + c;
            unsigned dst = (unsigned)(uintptr_t)&Q_s[buf][r * QP + c];
            asm volatile("global_load_async_to_lds_b128 %0, %1, off"
                         :: "v"(dst), "v"(src) : "memory");
        }
    };

    // Kick off chunk 0 immediately; it overlaps all of phases 1-4.
    issue_chunk(0, 0);

    if (tid < 15)    G_s[tid] = 0.f;
    if (tid < NSHOT) { pnorm_s[tid] = 0.f; mdot_s[tid] = 0.f; }

    // ---- Phase 1: class mean + centered shots Rc (one b128 pass, d0 = tid*4) ----
    const int d0 = tid * 4;
    {
        f4 v[NSHOT];
        f4 sum = {0.f, 0.f, 0.f, 0.f};
        #pragma unroll
        for (int s = 0; s < NSHOT; ++s) {
            f4 sv = *(const f4*)(support + (size_t)((t * NWAY + w) * NSHOT + s) * D_ + d0);
            f4 nv = *(const f4*)(noise   + (size_t)((w * T_ + t) * NSHOT + s) * D_ + d0);
            v[s] = sv + nv;
            sum += v[s];
        }
        f4 m = sum * 0.2f;
        *(f4*)&mean_s[d0] = m;
        #pragma unroll
        for (int s = 0; s < NSHOT; ++s) *(f4*)&Rc_s[s * RCP + d0] = v[s] - m;
    }
    __syncthreads();

    // ---- Phase 2: Gram G = Rc^T Rc (15 unique entries) ----
    {
        f4 r[NSHOT];
        #pragma unroll
        for (int s = 0; s < NSHOT; ++s) r[s] = *(const f4*)&Rc_s[s * RCP + d0];
        int idx = 0;
        #pragma unroll
        for (int a = 0; a < NSHOT; ++a)
            #pragma unroll
            for (int b = a; b < NSHOT; ++b) {
                f4 pr = r[a] * r[b];
                atomicAdd(&G_s[idx++], pr.x + pr.y + pr.z + pr.w);   // ds_add_f32
            }
    }
    __syncthreads();

    // ---- Phase 3: M = (lam*I + G)^-1 G  (exact replacement of the SVD shrinkage) ----
    if (tid == 0) {
        float G[5][5], A[5][5], X[5][5];
        int idx = 0;
        for (int a = 0; a < 5; ++a)
            for (int b = a; b < 5; ++b) { G[a][b] = G_s[idx]; G[b][a] = G_s[idx]; ++idx; }
        for (int a = 0; a < 5; ++a)
            for (int b = 0; b < 5; ++b) {
                A[a][b] = G[a][b] + (a == b ? LAMR : 0.f);
                X[a][b] = G[a][b];
            }
        for (int k = 0; k < 5; ++k) {           // Gauss-Jordan, SPD diag>=lam: no pivot
            float inv = 1.f / A[k][k];
            for (int j = 0; j < 5; ++j) { A[k][j] *= inv; X[k][j] *= inv; }
            for (int i = 0; i < 5; ++i) if (i != k) {
                float f = A[i][k];
                for (int j = 0; j < 5; ++j) { A[i][j] -= f * A[k][j]; X[i][j] -= f * X[k][j]; }
            }
        }
        for (int a = 0; a < 5; ++a)
            for (int b = 0; b < 5; ++b) M_s[a][b] = X[a][b];
    }
    __syncthreads();

    // ---- Phase 4: P = Rc*M, ||p||^2, mean.p ----
    {
        f4 r[NSHOT];
        #pragma unroll
        for (int s = 0; s < NSHOT; ++s) r[s] = *(const f4*)&Rc_s[s * RCP + d0];
        f4 mn = *(const f4*)&mean_s[d0];
        #pragma unroll
        for (int s2 = 0; s2 < NSHOT; ++s2) {
            f4 p = {0.f, 0.f, 0.f, 0.f};
            #pragma unroll
            for (int s1 = 0; s1 < NSHOT; ++s1) p += r[s1] * M_s[s1][s2];
            *(f4*)&P_s[s2 * RCP + d0] = p;
            f4 pp = p * p;
            f4 mp = mn * p;
            atomicAdd(&pnorm_s[s2], pp.x + pp.y + pp.z + pp.w);
            atomicAdd(&mdot_s[s2],  mp.x + mp.y + mp.z + mp.w);
        }
    }
    // ---- qnorm: one query row per wave, float4 lane-parallel, shuffle reduce ----
    for (int q = wave; q < NQ; q += 8) {
        const float* qr = qbase + (size_t)q * D_;
        float part = 0.f;
        for (int i = lane * 4; i < D_; i += 128) {
            f4 x = *(const f4*)(qr + i) - *(const f4*)&mean_s[i];
            f4 xx = x * x;
            part += xx.x + xx.y + xx.z + xx.w;
        }
        #pragma unroll
        for (int off = 16; off > 0; off >>= 1)
            part += __shfl_xor(part, off, 32);
        if (lane == 0) qnorm_s[q] = part;
    }
    __syncthreads();

    // ---- Phase 5: raw cross terms Q(75x1024) * P^T via f32 WMMA, double-buffered ----
    v8f acc = {};
    const int half = lane >> 4;                       // K-half within the 16x16x4 fragment
    const int l    = lane & 15;                       // M / N index within half
    const int arow = wave * 16 + l;                   // A row (query)
    const int brow = (l < NSHOT) ? l : (NSHOT - 1);   // clamp B rows 5..15 -> 4:
                                                      // finite garbage in D cols 5..15,
                                                      // which are never stored.
    int buf = 0;
    for (int kc = 0; kc < D_; kc += KC, buf ^= 1) {
        asm volatile("s_wait_asynccnt 0x0" ::: "memory");  // own copies into Q_s[buf] done
        __syncthreads();                                   // all copies done; prev reads done
        if (kc + KC < D_) issue_chunk(buf ^ 1, kc + KC);   // prefetch next tile
        if (wave < 5) {
            const float* arp = &Q_s[buf][arow * QP];
            const float* brp = &P_s[brow * RCP + kc];
            #pragma unroll
            for (int kk = 0; kk < KC; kk += 4) {
                const int c0 = kk + half * 2;              // even -> 8B aligned pairs
                v2f a = *(const v2f*)(arp + c0);           // unconditional ds_load_b64
                v2f b = *(const v2f*)(brp + c0);           // unconditional ds_load_b64
                acc = __builtin_amdgcn_wmma_f32_16x16x4_f32(
                        false, a, false, b, (short)0, acc, false, false);
            }
        }
    }

    // ---- Phase 6: dist = ||q-mean||^2 + ||p||^2 - 2*(q.p - mean.p); logits=-dist/d ----
    if (wave < 5 && l < NSHOT) {
        const int s = l;
        #pragma unroll
        for (int r = 0; r < 8; ++r) {
            int q = wave * 16 + half * 8 + r;   // C/D layout: VGPR r -> M=r (+8 hi half)
            if (q < NQ) {
                float dot  = acc[r] - mdot_s[s];
                float dist = qnorm_s[q] + pnorm_s[s] - 2.f * dot;
                out[(size_t)(t * NQ + q) * (NWAY * NSHOT) + w * NSHOT + s] =
                    -dist * (1.f / (float)D_);
            }
        }
    }
}

extern "C" void kernel_launch(void* const* d_in, const int* in_sizes, int n_in,
                              void* d_out, int out_size, void* d_ws, size_t ws_size,
                              hipStream_t stream) {
    (void)in_sizes; (void)n_in; (void)out_size; (void)d_ws; (void)ws_size;
    const float* query   = (const float*)d_in[0];
    const float* support = (const float*)d_in[1];
    const float* noise   = (const float*)d_in[2];
    float* out = (float*)d_out;
    exemplar_head_kernel<<<dim3(T_ * NWAY), dim3(256), 0, stream>>>(query, support, noise, out);
}